// identity_block3_dconv_10840497455945
// MI455X (gfx1250) — compile-verified
//
#include <hip/hip_runtime.h>

// identity_block3_dconv for MI455X (gfx1250, wave32, WMMA).
// All GEMMs use V_WMMA_F32_16X16X4_F32 (exact fp32).
// Block tile 128x128x16, wave tile 32x64 (8 x v8f accumulators),
// register-staged software pipeline for global->LDS.
// Workspace requirement: 2 * 32*256*784 * 4 B = ~51.4 MB.

typedef float v2f __attribute__((ext_vector_type(2)));
typedef float v8f __attribute__((ext_vector_type(8)));

#define HW 784
#define IMW 28

__device__ __forceinline__ v8f wmma4(v2f a, v2f b, v8f c) {
  return __builtin_amdgcn_wmma_f32_16x16x4_f32(false, a, false, b, (short)0, c,
                                               false, false);
}

// One 16-deep K step for a 32x64 wave tile from LDS tiles.
__device__ __forceinline__ void mma_tile(const float* a_lds, const float* b_lds,
                                         int wm2, int wn2, int lo, int hi,
                                         v8f acc[2][4]) {
#pragma unroll
  for (int k0 = 0; k0 < 16; k0 += 4) {
    v2f a0 = *(const v2f*)&a_lds[(wm2 * 32 + lo) * 18 + k0 + 2 * hi];
    v2f a1 = *(const v2f*)&a_lds[(wm2 * 32 + 16 + lo) * 18 + k0 + 2 * hi];
    int kp = (k0 >> 1) + hi;
#pragma unroll
    for (int j = 0; j < 4; ++j) {
      v2f bf = *(const v2f*)&b_lds[(kp * 144 + wn2 * 64 + j * 16 + lo) * 2];
      acc[0][j] = wmma4(a0, bf, acc[0][j]);
      acc[1][j] = wmma4(a1, bf, acc[1][j]);
    }
  }
}

// ---------------------------------------------------------------------------
// GEMM1: Y1[b] (256 x 784) = relu( W1 (256 x 1024) @ X[b] (1024 x 784) )
// grid (7, 2, 32), 256 threads.
// ---------------------------------------------------------------------------
__global__ __launch_bounds__(256)
void gemm1_kernel(const float* __restrict__ x, const float* __restrict__ w1,
                  float* __restrict__ y1) {
  __shared__ float a_lds[128 * 18];
  __shared__ float b_lds[8 * 144 * 2];

  const int tid = threadIdx.x;
  const int lane = tid & 31, wave = tid >> 5;
  const int wm2 = wave & 3, wn2 = wave >> 2;
  const int lo = lane & 15, hi = lane >> 4;
  const int nb = blockIdx.x * 128;
  const int mb = blockIdx.y * 128;
  const int b  = blockIdx.z;
  const float* Bsrc = x + (size_t)b * 1024 * HW;
  float* Out = y1 + (size_t)b * 256 * HW;
  const bool fastN = (nb + 128 <= HW);

  const int ar = tid >> 1;            // 0..127
  const int akq = (tid & 1) * 8;      // 0 or 8
  const int bk0 = tid >> 5;           // 0..7
  const int bn4 = (tid & 31) * 4;     // 0..124

  v8f zero = {};
  v8f acc[2][4];
#pragma unroll
  for (int mi = 0; mi < 2; ++mi)
#pragma unroll
    for (int j = 0; j < 4; ++j) acc[mi][j] = zero;

  float aR[8], bR[8];

  auto loadA = [&](int kt) {
    const float4 v0 = *(const float4*)&w1[(size_t)(mb + ar) * 1024 + kt + akq];
    const float4 v1 = *(const float4*)&w1[(size_t)(mb + ar) * 1024 + kt + akq + 4];
    aR[0] = v0.x; aR[1] = v0.y; aR[2] = v0.z; aR[3] = v0.w;
    aR[4] = v1.x; aR[5] = v1.y; aR[6] = v1.z; aR[7] = v1.w;
  };
  auto loadB = [&](int kt) {
#pragma unroll
    for (int rep = 0; rep < 2; ++rep) {
      int k = bk0 + rep * 8;
      if (fastN) {
        const float4 v = *(const float4*)&Bsrc[(size_t)(kt + k) * HW + nb + bn4];
        bR[rep * 4 + 0] = v.x; bR[rep * 4 + 1] = v.y;
        bR[rep * 4 + 2] = v.z; bR[rep * 4 + 3] = v.w;
      } else {
#pragma unroll
        for (int j = 0; j < 4; ++j) {
          int col = nb + bn4 + j;
          bR[rep * 4 + j] = (col < HW) ? Bsrc[(size_t)(kt + k) * HW + col] : 0.f;
        }
      }
    }
  };
  auto stage = [&]() {
    float* ad = &a_lds[ar * 18 + akq];
#pragma unroll
    for (int i = 0; i < 8; ++i) ad[i] = aR[i];
#pragma unroll
    for (int rep = 0; rep < 2; ++rep) {
      int k = bk0 + rep * 8;
      int kp = k >> 1, kb = k & 1;
#pragma unroll
      for (int j = 0; j < 4; ++j)
        b_lds[(kp * 144 + bn4 + j) * 2 + kb] = bR[rep * 4 + j];
    }
  };

  loadA(0); loadB(0);
  for (int kt = 0; kt < 1024; kt += 16) {
    stage();
    __syncthreads();
    if (kt + 16 < 1024) { loadA(kt + 16); loadB(kt + 16); }
    mma_tile(a_lds, b_lds, wm2, wn2, lo, hi, acc);
    __syncthreads();
  }

#pragma unroll
  for (int j = 0; j < 4; ++j) {
    int ng = nb + wn2 * 64 + j * 16 + lo;
    if (ng < HW) {
#pragma unroll
      for (int mi = 0; mi < 2; ++mi)
#pragma unroll
        for (int i = 0; i < 8; ++i) {
          int mg = mb + wm2 * 32 + mi * 16 + i + 8 * hi;
          Out[(size_t)mg * HW + ng] = fmaxf(acc[mi][j][i], 0.f);
        }
    }
  }
}

// ---------------------------------------------------------------------------
// Per-channel spatial gather: Ys[b,c,s] = Y1[b,c,perm[c,s]]
// ---------------------------------------------------------------------------
__global__ __launch_bounds__(256)
void shuffle_kernel(const float* __restrict__ y1, const int* __restrict__ perm,
                    float* __restrict__ ys) {
  int gid = blockIdx.x * 256 + threadIdx.x;
  if (gid < 32 * 256 * HW) {
    int s = gid % HW;
    int rowbase = gid - s;
    int c = (gid / HW) & 255;
    ys[gid] = y1[rowbase + perm[c * HW + s]];
  }
}

// ---------------------------------------------------------------------------
// GEMM2: 3x3 conv as implicit-im2col GEMM, K = 9 taps * 256 ci = 2304.
// k = tap*256 + ci: every 16-K tile has one fixed tap -> contiguous masked
// B loads. grid (7, 2, 32).
// ---------------------------------------------------------------------------
__global__ __launch_bounds__(256)
void gemm2_kernel(const float* __restrict__ ys, const float* __restrict__ wd,
                  float* __restrict__ y2) {
  __shared__ float a_lds[128 * 18];
  __shared__ float b_lds[8 * 144 * 2];

  const int tid = threadIdx.x;
  const int lane = tid & 31, wave = tid >> 5;
  const int wm2 = wave & 3, wn2 = wave >> 2;
  const int lo = lane & 15, hi = lane >> 4;
  const int nb = blockIdx.x * 128;
  const int mb = blockIdx.y * 128;
  const int b  = blockIdx.z;
  const float* Bsrc = ys + (size_t)b * 256 * HW;
  float* Out = y2 + (size_t)b * 256 * HW;

  const int ar = tid >> 1;
  const int akq = (tid & 1) * 8;
  const int bk0 = tid >> 5;
  const int bn4 = (tid & 31) * 4;

  v8f zero = {};
  v8f acc[2][4];
#pragma unroll
  for (int mi = 0; mi < 2; ++mi)
#pragma unroll
    for (int j = 0; j < 4; ++j) acc[mi][j] = zero;

  float aR[8], bR[8];

  auto loadA = [&](int kt) {
    int tap = kt >> 8;
    int ci0 = (kt & 255) + akq;
    const float* wrow = &wd[((size_t)(mb + ar) * 256 + ci0) * 9 + tap];
#pragma unroll
    for (int i = 0; i < 8; ++i) aR[i] = wrow[i * 9];
  };
  auto loadB = [&](int kt) {
    int tap = kt >> 8;
    int dh = tap / 3 - 1, dw = tap % 3 - 1;
    int cibase = kt & 255;
#pragma unroll
    for (int rep = 0; rep < 2; ++rep) {
      int k = bk0 + rep * 8;
      const float* row = &Bsrc[(size_t)(cibase + k) * HW];
#pragma unroll
      for (int j = 0; j < 4; ++j) {
        int s = nb + bn4 + j;
        float v = 0.f;
        if (s < HW) {
          int h = s / IMW + dh;
          int w = s % IMW + dw;
          if ((unsigned)h < (unsigned)IMW && (unsigned)w < (unsigned)IMW)
            v = row[h * IMW + w];
        }
        bR[rep * 4 + j] = v;
      }
    }
  };
  auto stage = [&]() {
    float* ad = &a_lds[ar * 18 + akq];
#pragma unroll
    for (int i = 0; i < 8; ++i) ad[i] = aR[i];
#pragma unroll
    for (int rep = 0; rep < 2; ++rep) {
      int k = bk0 + rep * 8;
      int kp = k >> 1, kb = k & 1;
#pragma unroll
      for (int j = 0; j < 4; ++j)
        b_lds[(kp * 144 + bn4 + j) * 2 + kb] = bR[rep * 4 + j];
    }
  };

  loadA(0); loadB(0);
  for (int kt = 0; kt < 2304; kt += 16) {
    stage();
    __syncthreads();
    if (kt + 16 < 2304) { loadA(kt + 16); loadB(kt + 16); }
    mma_tile(a_lds, b_lds, wm2, wn2, lo, hi, acc);
    __syncthreads();
  }

#pragma unroll
  for (int j = 0; j < 4; ++j) {
    int ng = nb + wn2 * 64 + j * 16 + lo;
    if (ng < HW) {
#pragma unroll
      for (int mi = 0; mi < 2; ++mi)
#pragma unroll
        for (int i = 0; i < 8; ++i) {
          int mg = mb + wm2 * 32 + mi * 16 + i + 8 * hi;
          Out[(size_t)mg * HW + ng] = fmaxf(acc[mi][j][i], 0.f);
        }
    }
  }
}

// ---------------------------------------------------------------------------
// GEMM3: out[b] = relu( W3 (1024x256) @ Y2[b] (256x784) + shuffled residual )
// grid (7, 8, 32).
// ---------------------------------------------------------------------------
__global__ __launch_bounds__(256)
void gemm3_kernel(const float* __restrict__ y2, const float* __restrict__ w3,
                  const float* __restrict__ x, const int* __restrict__ perm_res,
                  float* __restrict__ out) {
  __shared__ float a_lds[128 * 18];
  __shared__ float b_lds[8 * 144 * 2];

  const int tid = threadIdx.x;
  const int lane = tid & 31, wave = tid >> 5;
  const int wm2 = wave & 3, wn2 = wave >> 2;
  const int lo = lane & 15, hi = lane >> 4;
  const int nb = blockIdx.x * 128;
  const int mb = blockIdx.y * 128;
  const int b  = blockIdx.z;
  const float* Bsrc = y2 + (size_t)b * 256 * HW;
  const float* Xb   = x  + (size_t)b * 1024 * HW;
  float* Out = out + (size_t)b * 1024 * HW;
  const bool fastN = (nb + 128 <= HW);

  const int ar = tid >> 1;
  const int akq = (tid & 1) * 8;
  const int bk0 = tid >> 5;
  const int bn4 = (tid & 31) * 4;

  v8f zero = {};
  v8f acc[2][4];
#pragma unroll
  for (int mi = 0; mi < 2; ++mi)
#pragma unroll
    for (int j = 0; j < 4; ++j) acc[mi][j] = zero;

  float aR[8], bR[8];

  auto loadA = [&](int kt) {
    const float4 v0 = *(const float4*)&w3[(size_t)(mb + ar) * 256 + kt + akq];
    const float4 v1 = *(const float4*)&w3[(size_t)(mb + ar) * 256 + kt + akq + 4];
    aR[0] = v0.x; aR[1] = v0.y; aR[2] = v0.z; aR[3] = v0.w;
    aR[4] = v1.x; aR[5] = v1.y; aR[6] = v1.z; aR[7] = v1.w;
  };
  auto loadB = [&](int kt) {
#pragma unroll
    for (int rep = 0; rep < 2; ++rep) {
      int k = bk0 + rep * 8;
      if (fastN) {
        const float4 v = *(const float4*)&Bsrc[(size_t)(kt + k) * HW + nb + bn4];
        bR[rep * 4 + 0] = v.x; bR[rep * 4 + 1] = v.y;
        bR[rep * 4 + 2] = v.z; bR[rep * 4 + 3] = v.w;
      } else {
#pragma unroll
        for (int j = 0; j < 4; ++j) {
          int col = nb + bn4 + j;
          bR[rep * 4 + j] = (col < HW) ? Bsrc[(size_t)(kt + k) * HW + col] : 0.f;
        }
      }
    }
  };
  auto stage = [&]() {
    float* ad = &a_lds[ar * 18 + akq];
#pragma unroll
    for (int i = 0; i < 8; ++i) ad[i] = aR[i];
#pragma unroll
    for (int rep = 0; rep < 2; ++rep) {
      int k = bk0 + rep * 8;
      int kp = k >> 1, kb = k & 1;
#pragma unroll
      for (int j = 0; j < 4; ++j)
        b_lds[(kp * 144 + bn4 + j) * 2 + kb] = bR[rep * 4 + j];
    }
  };

  loadA(0); loadB(0);
  for (int kt = 0; kt < 256; kt += 16) {
    stage();
    __syncthreads();
    if (kt + 16 < 256) { loadA(kt + 16); loadB(kt + 16); }
    mma_tile(a_lds, b_lds, wm2, wn2, lo, hi, acc);
    __syncthreads();
  }

  // Epilogue: shuffled residual + relu
#pragma unroll
  for (int j = 0; j < 4; ++j) {
    int ng = nb + wn2 * 64 + j * 16 + lo;
    if (ng < HW) {
#pragma unroll
      for (int mi = 0; mi < 2; ++mi)
#pragma unroll
        for (int i = 0; i < 8; ++i) {
          int mg = mb + wm2 * 32 + mi * 16 + i + 8 * hi;
          float r = Xb[(size_t)mg * HW + perm_res[(size_t)mg * HW + ng]];
          Out[(size_t)mg * HW + ng] = fmaxf(acc[mi][j][i] + r, 0.f);
        }
    }
  }
}

// ---------------------------------------------------------------------------
extern "C" void kernel_launch(void* const* d_in, const int* in_sizes, int n_in,
                              void* d_out, int out_size, void* d_ws, size_t ws_size,
                              hipStream_t stream) {
  (void)in_sizes; (void)n_in; (void)out_size; (void)ws_size;
  const float* x          = (const float*)d_in[0];
  const float* w1         = (const float*)d_in[1];
  const float* wd         = (const float*)d_in[2];
  const float* w3         = (const float*)d_in[3];
  const int*   perm_dconv = (const int*)d_in[4];
  const int*   perm_res   = (const int*)d_in[5];
  float* out = (float*)d_out;

  float* y1 = (float*)d_ws;                      // 32*256*784 floats
  float* ys = y1 + (size_t)32 * 256 * HW;        // 32*256*784 floats
  float* y2 = y1;                                // reuse y1 buffer

  gemm1_kernel<<<dim3(7, 2, 32), 256, 0, stream>>>(x, w1, y1);
  shuffle_kernel<<<(32 * 256 * HW + 255) / 256, 256, 0, stream>>>(y1, perm_dconv, ys);
  gemm2_kernel<<<dim3(7, 2, 32), 256, 0, stream>>>(ys, wd, y2);
  gemm3_kernel<<<dim3(7, 8, 32), 256, 0, stream>>>(y2, w3, x, perm_res, out);
}